// SlicedReLUSelfAttention_32023276159754
// MI455X (gfx1250) — compile-verified
//
#include <hip/hip_runtime.h>
#include <hip/hip_bf16.h>
#include <math.h>

// Problem dims (fixed by the reference)
#define B_DIM 4
#define T_DIM 4096
#define E_DIM 1024
#define H_DIM 16
#define D_DIM 64
#define M_ROWS (B_DIM * T_DIM)   // 16384
#define N2_DIM (2 * T_DIM)       // 8192

typedef float v2f __attribute__((ext_vector_type(2)));
typedef float v8f __attribute__((ext_vector_type(8)));

// CDNA5 fp32 WMMA: D(16x16) = A(16x4) * B(4x16) + C
static __device__ __forceinline__ v8f wmma_f32(v2f a, v2f b, v8f c) {
  return __builtin_amdgcn_wmma_f32_16x16x4_f32(false, a, false, b, (short)0, c,
                                               false, false);
}

// ---------------------------------------------------------------------------
// Async global->LDS path (CDNA5 ASYNCcnt). Guarded so a toolchain without the
// builtins falls back to the synchronous staging path.
// Builtin signature (from clang diagnostics): the global-source parameter is
// addrspace(1) pointer to int __attribute__((vector_size(8/16))).
// ---------------------------------------------------------------------------
#if defined(__HIP_DEVICE_COMPILE__) &&                                        \
    __has_builtin(__builtin_amdgcn_global_load_async_to_lds_b64) &&           \
    __has_builtin(__builtin_amdgcn_global_load_async_to_lds_b128)
#define USE_ASYNC_LDS 1
#else
#define USE_ASYNC_LDS 0
#endif

#if USE_ASYNC_LDS
typedef int avi2 __attribute__((vector_size(8)));
typedef int avi4 __attribute__((vector_size(16)));
typedef __attribute__((address_space(1))) avi2* g_v2i_ptr;
typedef __attribute__((address_space(3))) avi2* l_v2i_ptr;
typedef __attribute__((address_space(1))) avi4* g_v4i_ptr;
typedef __attribute__((address_space(3))) avi4* l_v4i_ptr;

static __device__ __forceinline__ void async_g2l_b64(const float* g, float* l) {
  __builtin_amdgcn_global_load_async_to_lds_b64((g_v2i_ptr)(g), (l_v2i_ptr)(l),
                                                0, 0);
}
static __device__ __forceinline__ void async_g2l_b128(const float* g, float* l) {
  __builtin_amdgcn_global_load_async_to_lds_b128((g_v4i_ptr)(g), (l_v4i_ptr)(l),
                                                 0, 0);
}
static __device__ __forceinline__ void wait_async0() {
#if __has_builtin(__builtin_amdgcn_s_wait_asynccnt)
  __builtin_amdgcn_s_wait_asynccnt(0);
#else
  asm volatile("s_wait_asynccnt 0x0" ::: "memory");
#endif
}
#endif

// ---------------------------------------------------------------------------
// K1: Y = X @ W + bias  for Wq/Wk/Wv (selected by blockIdx.z)
//     X: (16384,1024)  W: (1024,1024)  Y: (16384,1024)
//     Workgroup tile 64x128, 8 waves, each wave 2x2 tiles of 16x16.
// ---------------------------------------------------------------------------
__global__ __launch_bounds__(256) void qkv_gemm_kernel(
    const float* __restrict__ X,
    const float* __restrict__ Wq, const float* __restrict__ bq,
    const float* __restrict__ Wk, const float* __restrict__ bk,
    const float* __restrict__ Wv, const float* __restrict__ bv,
    float* __restrict__ Oq, float* __restrict__ Ok, float* __restrict__ Ov)
{
  const float* W; const float* bias; float* O;
  if (blockIdx.z == 0)      { W = Wq; bias = bq; O = Oq; }
  else if (blockIdx.z == 1) { W = Wk; bias = bk; O = Ok; }
  else                      { W = Wv; bias = bv; O = Ov; }

  __shared__ float As[64][34];    // 64 rows x 32 k (pad 34: conflict-free frag reads)
  __shared__ float Bs[32][136];   // 32 k x 128 cols (pad 136)

  const int tid  = threadIdx.x;
  const int lane = tid & 31;
  const int wave = tid >> 5;
  const int lm   = lane & 15;
  const int lh   = lane >> 4;
  const int wrow = wave >> 2;     // 0..1
  const int wcol = wave & 3;      // 0..3
  const int rowbase = blockIdx.y * 64;
  const int colbase = blockIdx.x * 128;

  v8f acc[2][2];
  const v8f vzero = {0.f,0.f,0.f,0.f,0.f,0.f,0.f,0.f};
#pragma unroll
  for (int r = 0; r < 2; ++r)
#pragma unroll
    for (int s = 0; s < 2; ++s) acc[r][s] = vzero;

  for (int ks = 0; ks < E_DIM; ks += 32) {
    __syncthreads();
#if USE_ASYNC_LDS
#pragma unroll
    for (int it = 0; it < 4; ++it) {           // A tile: 1024 x b64
      int idx = tid + it * 256;
      int r = idx >> 4, c2 = (idx & 15) * 2;
      async_g2l_b64(&X[(size_t)(rowbase + r) * E_DIM + (ks + c2)], &As[r][c2]);
    }
#pragma unroll
    for (int it = 0; it < 4; ++it) {           // B tile: 1024 x b128
      int idx = tid + it * 256;
      int r = idx >> 5, c4 = (idx & 31) * 4;
      async_g2l_b128(&W[(size_t)(ks + r) * E_DIM + (colbase + c4)], &Bs[r][c4]);
    }
    if (ks + 32 < E_DIM) {                     // L2 prefetch of next stage
      __builtin_prefetch(&X[(size_t)(rowbase + (tid >> 2)) * E_DIM +
                            (ks + 32) + (tid & 3) * 8], 0, 1);
      __builtin_prefetch(&W[(size_t)(ks + 32 + (tid >> 3)) * E_DIM + colbase +
                            (tid & 7) * 16], 0, 1);
    }
    wait_async0();
#else
#pragma unroll
    for (int it = 0; it < 8; ++it) {           // A tile: 2048 floats
      int idx = tid + it * 256;
      int r = idx >> 5, c = idx & 31;
      As[r][c] = X[(size_t)(rowbase + r) * E_DIM + (ks + c)];
    }
#pragma unroll
    for (int it = 0; it < 16; ++it) {          // B tile: 4096 floats
      int idx = tid + it * 256;
      int r = idx >> 7, c = idx & 127;
      Bs[r][c] = W[(size_t)(ks + r) * E_DIM + (colbase + c)];
    }
#endif
    __syncthreads();

#pragma unroll
    for (int kk = 0; kk < 32; kk += 4) {
      const int kb = kk + 2 * lh;
      v2f a[2], bf[2];
#pragma unroll
      for (int r = 0; r < 2; ++r) {
        int mr = wrow * 32 + r * 16 + lm;
        a[r].x = As[mr][kb];
        a[r].y = As[mr][kb + 1];
      }
#pragma unroll
      for (int s = 0; s < 2; ++s) {
        int nc = wcol * 32 + s * 16 + lm;
        bf[s].x = Bs[kb][nc];
        bf[s].y = Bs[kb + 1][nc];
      }
#pragma unroll
      for (int r = 0; r < 2; ++r)
#pragma unroll
        for (int s = 0; s < 2; ++s)
          acc[r][s] = wmma_f32(a[r], bf[s], acc[r][s]);
    }
  }

#pragma unroll
  for (int r = 0; r < 2; ++r) {
#pragma unroll
    for (int s = 0; s < 2; ++s) {
      int col = colbase + wcol * 32 + s * 16 + lm;
      float bv_ = bias[col];
#pragma unroll
      for (int j = 0; j < 8; ++j) {
        int row = rowbase + wrow * 32 + r * 16 + j + 8 * lh;
        O[(size_t)row * E_DIM + col] = acc[r][s][j] + bv_;
      }
    }
  }
}

// ---------------------------------------------------------------------------
// K2: fused projection:  P(b,t2,0:16) = gelu( Xg @ W1 + b1 ) @ W2 + b2
//     Xg is the scrambled-reshape gather of q (or k):
//       Xg[t2,kk] = Q[b, ((t2&255)<<4) + (kk>>6), ((t2>>8)<<6) + (kk&63)]
//     One workgroup per 64 t2-rows; loops 8 x 128-wide M1 blocks, keeping the
//     64x16 result in registers across the whole contraction (deterministic).
// ---------------------------------------------------------------------------
__global__ __launch_bounds__(256) void proj_kernel(
    const float* __restrict__ Q,
    const float* __restrict__ W1, const float* __restrict__ b1,
    const float* __restrict__ W2, const float* __restrict__ b2,
    float* __restrict__ P)
{
  __shared__ float As[64][34];
  __shared__ float Bs[32][136];
  __shared__ float Ms[64][133];   // M1 tile re-striping buffer (pad 133)

  const int tid  = threadIdx.x;
  const int lane = tid & 31;
  const int wave = tid >> 5;
  const int lm   = lane & 15;
  const int lh   = lane >> 4;
  const int wrow = wave >> 2;
  const int wcol = wave & 3;
  const int b       = blockIdx.y;
  const int rowbase = blockIdx.x * 64;

  const v8f vzero = {0.f,0.f,0.f,0.f,0.f,0.f,0.f,0.f};
  v8f acc2 = vzero;   // 16x16 output accumulator (waves 0..3 only)

  for (int nb = 0; nb < 8; ++nb) {
    v8f acc[2][2];
#pragma unroll
    for (int r = 0; r < 2; ++r)
#pragma unroll
      for (int s = 0; s < 2; ++s) acc[r][s] = vzero;

    // ---- phase A: M1 tile (64 x 128) = gelu(Xg @ W1 + b1) ----
    for (int ks = 0; ks < E_DIM; ks += 32) {
      __syncthreads();
#if USE_ASYNC_LDS
#pragma unroll
      for (int it = 0; it < 4; ++it) {         // gathered A tile: 1024 x b64
        int idx = tid + it * 256;
        int r = idx >> 4, c2 = (idx & 15) * 2;
        int t2  = rowbase + r;
        int kkg = ks + c2;                     // kkg, kkg+1 share (kkg>>6)
        int trow = ((t2 & 255) << 4) + (kkg >> 6);
        int qcol = ((t2 >> 8) << 6) + (kkg & 63);
        async_g2l_b64(&Q[(size_t)(b * T_DIM + trow) * E_DIM + qcol], &As[r][c2]);
      }
#pragma unroll
      for (int it = 0; it < 4; ++it) {         // W1 tile: 1024 x b128
        int idx = tid + it * 256;
        int r = idx >> 5, c4 = (idx & 31) * 4;
        async_g2l_b128(&W1[(size_t)(ks + r) * E_DIM + (nb * 128 + c4)],
                       &Bs[r][c4]);
      }
      wait_async0();
#else
#pragma unroll
      for (int it = 0; it < 8; ++it) {
        int idx = tid + it * 256;
        int r = idx >> 5, c = idx & 31;
        int t2  = rowbase + r;
        int kkg = ks + c;
        int trow = ((t2 & 255) << 4) + (kkg >> 6);
        int qcol = ((t2 >> 8) << 6) + (kkg & 63);
        As[r][c] = Q[(size_t)(b * T_DIM + trow) * E_DIM + qcol];
      }
#pragma unroll
      for (int it = 0; it < 16; ++it) {
        int idx = tid + it * 256;
        int r = idx >> 7, c = idx & 127;
        Bs[r][c] = W1[(size_t)(ks + r) * E_DIM + (nb * 128 + c)];
      }
#endif
      __syncthreads();

#pragma unroll
      for (int kk = 0; kk < 32; kk += 4) {
        const int kb = kk + 2 * lh;
        v2f a[2], bf[2];
#pragma unroll
        for (int r = 0; r < 2; ++r) {
          int mr = wrow * 32 + r * 16 + lm;
          a[r].x = As[mr][kb];
          a[r].y = As[mr][kb + 1];
        }
#pragma unroll
        for (int s = 0; s < 2; ++s) {
          int nc = wcol * 32 + s * 16 + lm;
          bf[s].x = Bs[kb][nc];
          bf[s].y = Bs[kb + 1][nc];
        }
#pragma unroll
        for (int r = 0; r < 2; ++r)
#pragma unroll
          for (int s = 0; s < 2; ++s)
            acc[r][s] = wmma_f32(a[r], bf[s], acc[r][s]);
      }
    }

    // bias + exact GELU, write C-layout -> LDS for re-striping
#pragma unroll
    for (int r = 0; r < 2; ++r) {
#pragma unroll
      for (int s = 0; s < 2; ++s) {
        int cl = wcol * 32 + s * 16 + lm;
        float bb = b1[nb * 128 + cl];
#pragma unroll
        for (int j = 0; j < 8; ++j) {
          float x = acc[r][s][j] + bb;
          x = 0.5f * x * (1.0f + erff(x * 0.70710678118654752f));
          Ms[wrow * 32 + r * 16 + j + 8 * lh][cl] = x;
        }
      }
    }
    __syncthreads();

    // ---- phase B: acc2 += M1tile(64x128) @ W2[nb*128 : nb*128+128, 0:16] ----
    if (wave < 4) {
      for (int kk = 0; kk < 128; kk += 4) {
        const int kb = kk + 2 * lh;
        v2f a, bw;
        a.x = Ms[wave * 16 + lm][kb];
        a.y = Ms[wave * 16 + lm][kb + 1];
        bw.x = W2[(size_t)(nb * 128 + kb) * H_DIM + lm];
        bw.y = W2[(size_t)(nb * 128 + kb + 1) * H_DIM + lm];
        acc2 = wmma_f32(a, bw, acc2);
      }
    }
    __syncthreads();
  }

  if (wave < 4) {
    float bias2 = b2[lm];
#pragma unroll
    for (int j = 0; j < 8; ++j) {
      int row = rowbase + wave * 16 + j + 8 * lh;
      P[(size_t)(b * T_DIM + row) * H_DIM + lm] = acc2[j] + bias2;
    }
  }
}

// ---------------------------------------------------------------------------
// K3: per-(b,h) mean/var of q_proj & k_proj columns over T; store mean_q,
//     mean_k and rstd = 1/sqrt(0.5*(var_q+var_k)+1e-6).
// ---------------------------------------------------------------------------
__global__ __launch_bounds__(256) void stats_kernel(
    const float* __restrict__ QP, const float* __restrict__ KP,
    float* __restrict__ MQ, float* __restrict__ MK, float* __restrict__ RSTD)
{
  const int g = blockIdx.x, b = g >> 4, h = g & 15;
  const int tid = threadIdx.x;
  float sq = 0.f, sq2 = 0.f, sk = 0.f, sk2 = 0.f;
  for (int i = tid; i < T_DIM; i += 256) {
    float q = QP[(size_t)(b * T_DIM + i) * H_DIM + h];
    float k = KP[(size_t)(b * T_DIM + i) * H_DIM + h];
    sq += q; sq2 += q * q; sk += k; sk2 += k * k;
  }
  __shared__ float red[4][256];
  red[0][tid] = sq; red[1][tid] = sq2; red[2][tid] = sk; red[3][tid] = sk2;
  __syncthreads();
  for (int off = 128; off > 0; off >>= 1) {
    if (tid < off)
      for (int a = 0; a < 4; ++a) red[a][tid] += red[a][tid + off];
    __syncthreads();
  }
  if (tid == 0) {
    const float inv = 1.0f / (float)T_DIM;
    float mq = red[0][0] * inv, mk = red[2][0] * inv;
    float vq = red[1][0] * inv - mq * mq;
    float vk = red[3][0] * inv - mk * mk;
    float std = sqrtf(0.5f * (vq + vk) + 1e-6f);
    MQ[g] = mq; MK[g] = mk; RSTD[g] = 1.0f / std;
  }
}

// ---------------------------------------------------------------------------
// K3b: mean of v over T per (b,h,d)
// ---------------------------------------------------------------------------
__global__ __launch_bounds__(256) void meanv_kernel(
    const float* __restrict__ V, float* __restrict__ MV)
{
  const int g = blockIdx.x, b = g >> 4, h = g & 15;
  const int tid = threadIdx.x, d = tid & 63, c = tid >> 6;
  float s = 0.f;
  for (int t = c * 1024; t < (c + 1) * 1024; ++t)
    s += V[(size_t)(b * T_DIM + t) * E_DIM + h * 64 + d];
  __shared__ float red[4][64];
  red[c][d] = s;
  __syncthreads();
  if (tid < 64)
    MV[g * 64 + tid] =
        (red[0][tid] + red[1][tid] + red[2][tid] + red[3][tid]) *
        (1.0f / (float)T_DIM);
}

// ---------------------------------------------------------------------------
// K4: per-(b,h) build Z (keys then queries, normalized) and bitonic-sort the
//     8192 (value, origin-index) pairs in LDS.
// ---------------------------------------------------------------------------
__global__ __launch_bounds__(1024) void sort_kernel(
    const float* __restrict__ QP, const float* __restrict__ KP,
    const float* __restrict__ MQ, const float* __restrict__ MK,
    const float* __restrict__ RSTD,
    float* __restrict__ ZS, int* __restrict__ SRC)
{
  __shared__ float sv[N2_DIM];
  __shared__ int   si[N2_DIM];
  const int g = blockIdx.x, b = g >> 4, h = g & 15;
  const int tid = threadIdx.x;
  const float mq = MQ[g], mk = MK[g], rs = RSTD[g];

  for (int i = tid; i < N2_DIM; i += 1024) {
    float z;
    if (i < T_DIM)
      z = (KP[(size_t)(b * T_DIM + i) * H_DIM + h] - mk) * rs;
    else
      z = (QP[(size_t)(b * T_DIM + (i - T_DIM)) * H_DIM + h] - mq) * rs;
    sv[i] = z; si[i] = i;
  }

  for (int size = 2; size <= N2_DIM; size <<= 1) {
    for (int stride = size >> 1; stride > 0; stride >>= 1) {
      __syncthreads();
      for (int t = tid; t < (N2_DIM / 2); t += 1024) {
        int i = ((t & ~(stride - 1)) << 1) | (t & (stride - 1));
        int j = i | stride;
        bool up = ((i & size) == 0);
        float vi = sv[i], vj = sv[j];
        if ((vi > vj) == up) {
          sv[i] = vj; sv[j] = vi;
          int ti = si[i]; si[i] = si[j]; si[j] = ti;
        }
      }
    }
  }
  __syncthreads();
  for (int i = tid; i < N2_DIM; i += 1024) {
    ZS[(size_t)g * N2_DIM + i]  = sv[i];
    SRC[(size_t)g * N2_DIM + i] = si[i];
  }
}

// ---------------------------------------------------------------------------
// K5: segmented inclusive scans over sorted order (4 segments x 2048 ranks,
//     64 dims across lanes), emit 2*ctx/max(norm,1e-4) at query slots.
// ---------------------------------------------------------------------------
__global__ __launch_bounds__(256) void scan_kernel(
    const float* __restrict__ V, const float* __restrict__ MV,
    const float* __restrict__ ZS, const int* __restrict__ SRC,
    float* __restrict__ OUT)
{
  const int g = blockIdx.x, b = g >> 4, h = g & 15;
  const int tid = threadIdx.x, d = tid & 63, sg = tid >> 6;
  const float* zsg = ZS + (size_t)g * N2_DIM;
  const int*  srcg = SRC + (size_t)g * N2_DIM;
  const float mv = MV[g * 64 + d];
  const int r0 = sg * 2048, r1 = r0 + 2048;

  // pass 1: segment totals
  float sP = 0.f, sPZ = 0.f, sW = 0.f, sWZ = 0.f;
  for (int r = r0; r < r1; ++r) {
    int i = srcg[r];
    float z = zsg[r];
    if (i < T_DIM) {
      float p = V[(size_t)(b * T_DIM + i) * E_DIM + h * 64 + d] - mv;
      sP += p; sPZ += p * z; sW += 1.f; sWZ += z;
    }
  }
  __shared__ float segP[4][64], segPZ[4][64], segW[4], segWZ[4];
  segP[sg][d] = sP; segPZ[sg][d] = sPZ;
  if (d == 0) { segW[sg] = sW; segWZ[sg] = sWZ; }
  __syncthreads();

  float cP = 0.f, cPZ = 0.f, cW = 0.f, cWZ = 0.f, totW = 0.f, totWZ = 0.f;
  for (int g2 = 0; g2 < 4; ++g2) {
    if (g2 < sg) {
      cP += segP[g2][d]; cPZ += segPZ[g2][d];
      cW += segW[g2];    cWZ += segWZ[g2];
    }
    totW += segW[g2]; totWZ += segWZ[g2];
  }

  // pass 2: scan + emit at query slots
  for (int r = r0; r < r1; ++r) {
    int i = srcg[r];
    float z = zsg[r];
    if (i < T_DIM) {
      float p = V[(size_t)(b * T_DIM + i) * E_DIM + h * 64 + d] - mv;
      cP += p; cPZ += p * z; cW += 1.f; cWZ += z;
    } else {
      float ctx  = cP * z - cPZ;
      float srel = cW * z - cWZ;
      float norm = 2.f * srel - (z * totW - totWZ);
      norm = fmaxf(norm, 1e-4f);
      int tq = i - T_DIM;
      OUT[(size_t)(b * T_DIM + tq) * E_DIM + h * 64 + d] = 2.f * ctx / norm;
    }
  }
}

// ---------------------------------------------------------------------------
// Workspace layout (floats)
// ---------------------------------------------------------------------------
static const size_t OFF_Q    = 0;
static const size_t OFF_K    = OFF_Q  + (size_t)M_ROWS * E_DIM;   // 16777216
static const size_t OFF_V    = OFF_K  + (size_t)M_ROWS * E_DIM;
static const size_t OFF_QP   = OFF_V  + (size_t)M_ROWS * E_DIM;
static const size_t OFF_KP   = OFF_QP + (size_t)M_ROWS * H_DIM;
static const size_t OFF_MQ   = OFF_KP + (size_t)M_ROWS * H_DIM;
static const size_t OFF_MK   = OFF_MQ + 64;
static const size_t OFF_RSTD = OFF_MK + 64;
static const size_t OFF_MV   = OFF_RSTD + 64;
static const size_t OFF_ZS   = OFF_MV + (size_t)B_DIM * H_DIM * D_DIM;
static const size_t OFF_SRC  = OFF_ZS + (size_t)B_DIM * H_DIM * N2_DIM;

extern "C" void kernel_launch(void* const* d_in, const int* in_sizes, int n_in,
                              void* d_out, int out_size, void* d_ws,
                              size_t ws_size, hipStream_t stream)
{
  (void)in_sizes; (void)n_in; (void)out_size; (void)ws_size;
  const float* X  = (const float*)d_in[0];
  const float* Wq = (const float*)d_in[1];
  const float* bq = (const float*)d_in[2];
  const float* Wk = (const float*)d_in[3];
  const float* bk = (const float*)d_in[4];
  const float* Wv = (const float*)d_in[5];
  const float* bv = (const float*)d_in[6];
  const float* W1 = (const float*)d_in[7];
  const float* b1 = (const float*)d_in[8];
  const float* W2 = (const float*)d_in[9];
  const float* b2 = (const float*)d_in[10];
  float* out = (float*)d_out;
  float* ws  = (float*)d_ws;

  float* q    = ws + OFF_Q;
  float* k    = ws + OFF_K;
  float* v    = ws + OFF_V;
  float* qp   = ws + OFF_QP;
  float* kp   = ws + OFF_KP;
  float* mq   = ws + OFF_MQ;
  float* mk   = ws + OFF_MK;
  float* rstd = ws + OFF_RSTD;
  float* mv   = ws + OFF_MV;
  float* zs   = ws + OFF_ZS;
  int*   srci = (int*)(ws + OFF_SRC);

  qkv_gemm_kernel<<<dim3(E_DIM / 128, M_ROWS / 64, 3), 256, 0, stream>>>(
      X, Wq, bq, Wk, bk, Wv, bv, q, k, v);
  proj_kernel<<<dim3(T_DIM / 64, B_DIM), 256, 0, stream>>>(q, W1, b1, W2, b2, qp);
  proj_kernel<<<dim3(T_DIM / 64, B_DIM), 256, 0, stream>>>(k, W1, b1, W2, b2, kp);
  stats_kernel<<<B_DIM * H_DIM, 256, 0, stream>>>(qp, kp, mq, mk, rstd);
  meanv_kernel<<<B_DIM * H_DIM, 256, 0, stream>>>(v, mv);
  sort_kernel<<<B_DIM * H_DIM, 1024, 0, stream>>>(qp, kp, mq, mk, rstd, zs, srci);
  scan_kernel<<<B_DIM * H_DIM, 256, 0, stream>>>(v, mv, zs, srci, out);
}